// SelfAttentionTransformer_65481071408156
// MI455X (gfx1250) — compile-verified
//
#include <hip/hip_runtime.h>
#include <math.h>

// Problem constants (from reference): B=4, L=2048, D=256, H=8, dk=32, F=1024
#define KL 2048
#define KD 256
#define KB 4
#define KF 1024
#define KH 8
#define KDK 32
#define LN_EPS 1e-5f
#define PI_F 3.14159265358979323846f

typedef float v2f __attribute__((ext_vector_type(2)));
typedef float v8f __attribute__((ext_vector_type(8)));

// D(16x16,f32) = A(16x4,f32) * B(4x16,f32) + C   (wave32 fragments)
//   A: row = lane&15, ks = 2*(lane>>4);  a = {A[row][ks], A[row][ks+1]}
//   B: col = lane&15, ks = 2*(lane>>4);  b = {B[ks][col], B[ks+1][col]}
//   C/D: element r -> C[r + 8*(lane>>4)][lane&15]
__device__ __forceinline__ v8f wmma4(v2f a, v2f b, v8f c) {
  return __builtin_amdgcn_wmma_f32_16x16x4_f32(
      /*neg_a=*/false, a, /*neg_b=*/false, b,
      /*c_mod=*/(short)0, c, /*reuse_a=*/false, /*reuse_b=*/false);
}

// wave32 all-lanes sum via lane-permute (no LDS, no barriers)
__device__ __forceinline__ float waveSum(float v) {
#pragma unroll
  for (int off = 16; off > 0; off >>= 1) v += __shfl_xor(v, off, 32);
  return v;
}

// ---------------------------------------------------------------------------
// 1) m[l][k] = mean over (b,h) of x[b][l][h*32+k]   -> [L,32]
// ---------------------------------------------------------------------------
__global__ void qat_mean(const float* __restrict__ x, float* __restrict__ m) {
  int idx = blockIdx.x * 256 + threadIdx.x;    // L*32 = 65536 threads
  int l = idx >> 5, k = idx & 31;
  float s = 0.f;
#pragma unroll
  for (int b = 0; b < KB; ++b)
#pragma unroll
    for (int hh = 0; hh < KH; ++hh)
      s += x[(size_t)(b * KL + l) * KD + hh * KDK + k];
  m[idx] = s * (1.f / (KB * KH));
}

// ---------------------------------------------------------------------------
// 1b) xm = mask o x  (hoists the key-mask out of the attention GEMM hot loop)
// ---------------------------------------------------------------------------
__global__ void qat_maskx(const float* __restrict__ x, const int* __restrict__ mask,
                          float* __restrict__ xm) {
  int idx = blockIdx.x * 256 + threadIdx.x;    // B*L*D threads
  int bl = idx >> 8;                           // b*L + l
  xm[idx] = mask[bl] ? x[idx] : 0.f;
}

// ---------------------------------------------------------------------------
// 2) E[i][j] = exp( sin^2( clip(m_i . m_j, -pi, pi) / 2 ) )  via WMMA f32
//    grid (L/16, L/128), block 256 (8 waves, one 16x16 tile each)
//    |ang/2| <= pi/2, so hardware v_sin_f32/v_exp_f32 are exact enough —
//    avoids the libm Payne-Hanek slow path entirely.
// ---------------------------------------------------------------------------
__global__ void qat_scores(const float* __restrict__ m, float* __restrict__ E) {
  int w = threadIdx.x >> 5, lane = threadIdx.x & 31;
  int rc = lane & 15;                  // A-row / B-col
  int kb = (lane >> 4) << 1;           // 0 or 2
  int half8 = (lane >> 4) << 3;        // 0 or 8
  int i0 = blockIdx.x * 16;
  int j0 = (blockIdx.y * 8 + w) * 16;
  v8f acc = {};
#pragma unroll
  for (int k0 = 0; k0 < KDK; k0 += 4) {
    v2f a = *(const v2f*)(m + (i0 + rc) * KDK + k0 + kb);   // m[i][k..]
    v2f b = *(const v2f*)(m + (j0 + rc) * KDK + k0 + kb);   // (m^T)[k][j] = m[j][k..]
    acc = wmma4(a, b, acc);
  }
#pragma unroll
  for (int r = 0; r < 8; ++r) {
    float dot = acc[r];
    float ang = fminf(fmaxf(dot, -PI_F), PI_F);
    float s = __sinf(0.5f * ang);                            // v_sin_f32
    s = s * s;                                               // score in [0,1]
    E[(size_t)(i0 + r + half8) * KL + (j0 + rc)] = __expf(s); // v_exp_f32
  }
}

// ---------------------------------------------------------------------------
// 3) rowsum[b][i] = sum_j E[i][j] * mask[b][j]   (shuffle + 8-way LDS combine)
// ---------------------------------------------------------------------------
__global__ void qat_rowsum(const float* __restrict__ E, const int* __restrict__ mask,
                           float* __restrict__ rowsum) {
  int bi = blockIdx.x;                 // b*L + i
  int b = bi / KL, i = bi - b * KL;
  int t = threadIdx.x;
  const float* Er = E + (size_t)i * KL;
  const int* mb = mask + b * KL;
  float s = 0.f;
  for (int j = t; j < KL; j += 256) s += mb[j] ? Er[j] : 0.f;
  s = waveSum(s);
  __shared__ float part[8];
  int wave = t >> 5, lane = t & 31;
  if (lane == 0) part[wave] = s;
  __syncthreads();
  if (t == 0) {
    float tot = 0.f;
#pragma unroll
    for (int k = 0; k < 8; ++k) tot += part[k];
    rowsum[bi] = tot;
  }
}

// ---------------------------------------------------------------------------
// 4) O[b] = E @ xm[b]  [2048x256, K=2048]   WMMA f32
//    grid (L/64, D/128, B), block 256 = 8 waves, each wave a 16x64 tile
// ---------------------------------------------------------------------------
__global__ void qat_attn_gemm(const float* __restrict__ E, const float* __restrict__ xm,
                              float* __restrict__ O) {
  int b = blockIdx.z;
  int w = threadIdx.x >> 5, lane = threadIdx.x & 31;
  int rc = lane & 15;
  int kb = (lane >> 4) << 1;
  int half8 = (lane >> 4) << 3;
  int i0 = blockIdx.x * 64 + (w >> 1) * 16;
  int nb = blockIdx.y * 128 + (w & 1) * 64;
  const float* xb = xm + (size_t)b * KL * KD;
  v8f acc[4] = {};
  for (int k0 = 0; k0 < KL; k0 += 4) {
    int k = k0 + kb;
    v2f a = *(const v2f*)(E + (size_t)(i0 + rc) * KL + k);
#pragma unroll
    for (int nt = 0; nt < 4; ++nt) {
      int n = nb + nt * 16 + rc;
      v2f bf;
      bf.x = xb[(size_t)k * KD + n];
      bf.y = xb[(size_t)(k + 1) * KD + n];
      acc[nt] = wmma4(a, bf, acc[nt]);
    }
  }
#pragma unroll
  for (int nt = 0; nt < 4; ++nt)
#pragma unroll
    for (int r = 0; r < 8; ++r)
      O[(size_t)b * KL * KD + (size_t)(i0 + r + half8) * KD + nb + nt * 16 + rc] = acc[nt][r];
}

// ---------------------------------------------------------------------------
// 5) h = LayerNorm(x + O/rowsum) * g1 + be1     one block per token row
// ---------------------------------------------------------------------------
__global__ void qat_ln1(const float* __restrict__ x, const float* __restrict__ O,
                        const float* __restrict__ rowsum, const float* __restrict__ g,
                        const float* __restrict__ be, float* __restrict__ h) {
  int rowId = blockIdx.x;              // b*L + l
  int t = threadIdx.x;                 // 0..255 == feature dim
  int wave = t >> 5, lane = t & 31;
  __shared__ float part[8];
  float inv = 1.0f / rowsum[rowId];
  float v = x[(size_t)rowId * KD + t] + O[(size_t)rowId * KD + t] * inv;

  float s = waveSum(v);
  if (lane == 0) part[wave] = s;
  __syncthreads();
  float tot = 0.f;
#pragma unroll
  for (int k = 0; k < 8; ++k) tot += part[k];
  float mu = tot * (1.0f / KD);
  float d = v - mu;

  float s2 = waveSum(d * d);
  __syncthreads();                     // protect part[] before rewrite
  if (lane == 0) part[wave] = s2;
  __syncthreads();
  float tot2 = 0.f;
#pragma unroll
  for (int k = 0; k < 8; ++k) tot2 += part[k];
  float rstd = rsqrtf(tot2 * (1.0f / KD) + LN_EPS);

  h[(size_t)rowId * KD + t] = d * rstd * g[t] + be[t];
}

// ---------------------------------------------------------------------------
// 6) hid = relu(h @ W1 + b1)   [8192x1024, K=256]   WMMA f32
//    grid (BL/32, F/256), block 256 = 8 waves (2 M-tiles x 4 N-tiles of 64)
// ---------------------------------------------------------------------------
__global__ void qat_ffn1(const float* __restrict__ h, const float* __restrict__ W1,
                         const float* __restrict__ b1, float* __restrict__ hid) {
  int w = threadIdx.x >> 5, lane = threadIdx.x & 31;
  int rc = lane & 15;
  int kb = (lane >> 4) << 1;
  int half8 = (lane >> 4) << 3;
  int i0 = blockIdx.x * 32 + (w >> 2) * 16;
  int nb = blockIdx.y * 256 + (w & 3) * 64;
  v8f acc[4] = {};
  for (int k0 = 0; k0 < KD; k0 += 4) {
    int k = k0 + kb;
    v2f a = *(const v2f*)(h + (size_t)(i0 + rc) * KD + k);
#pragma unroll
    for (int nt = 0; nt < 4; ++nt) {
      int n = nb + nt * 16 + rc;
      v2f bf;
      bf.x = W1[(size_t)k * KF + n];
      bf.y = W1[(size_t)(k + 1) * KF + n];
      acc[nt] = wmma4(a, bf, acc[nt]);
    }
  }
#pragma unroll
  for (int nt = 0; nt < 4; ++nt)
#pragma unroll
    for (int r = 0; r < 8; ++r) {
      int n = nb + nt * 16 + rc;
      float v = acc[nt][r] + b1[n];
      hid[(size_t)(i0 + r + half8) * KF + n] = fmaxf(v, 0.f);
    }
}

// ---------------------------------------------------------------------------
// 7) f2 = hid @ W2 + b2   [8192x256, K=1024]   WMMA f32
// ---------------------------------------------------------------------------
__global__ void qat_ffn2(const float* __restrict__ hid, const float* __restrict__ W2,
                         const float* __restrict__ b2, float* __restrict__ f2) {
  int w = threadIdx.x >> 5, lane = threadIdx.x & 31;
  int rc = lane & 15;
  int kb = (lane >> 4) << 1;
  int half8 = (lane >> 4) << 3;
  int i0 = blockIdx.x * 32 + (w >> 2) * 16;
  int nb = blockIdx.y * 256 + (w & 3) * 64;
  v8f acc[4] = {};
  for (int k0 = 0; k0 < KF; k0 += 4) {
    int k = k0 + kb;
    v2f a = *(const v2f*)(hid + (size_t)(i0 + rc) * KF + k);
#pragma unroll
    for (int nt = 0; nt < 4; ++nt) {
      int n = nb + nt * 16 + rc;
      v2f bf;
      bf.x = W2[(size_t)k * KD + n];
      bf.y = W2[(size_t)(k + 1) * KD + n];
      acc[nt] = wmma4(a, bf, acc[nt]);
    }
  }
#pragma unroll
  for (int nt = 0; nt < 4; ++nt)
#pragma unroll
    for (int r = 0; r < 8; ++r) {
      int n = nb + nt * 16 + rc;
      f2[(size_t)(i0 + r + half8) * KD + n] = acc[nt][r] + b2[n];
    }
}

// ---------------------------------------------------------------------------
// 8) out = LayerNorm(h + f2) * g2 + be2
// ---------------------------------------------------------------------------
__global__ void qat_ln2(const float* __restrict__ h, const float* __restrict__ f2,
                        const float* __restrict__ g, const float* __restrict__ be,
                        float* __restrict__ out) {
  int rowId = blockIdx.x;
  int t = threadIdx.x;
  int wave = t >> 5, lane = t & 31;
  __shared__ float part[8];
  float v = h[(size_t)rowId * KD + t] + f2[(size_t)rowId * KD + t];

  float s = waveSum(v);
  if (lane == 0) part[wave] = s;
  __syncthreads();
  float tot = 0.f;
#pragma unroll
  for (int k = 0; k < 8; ++k) tot += part[k];
  float mu = tot * (1.0f / KD);
  float d = v - mu;

  float s2 = waveSum(d * d);
  __syncthreads();
  if (lane == 0) part[wave] = s2;
  __syncthreads();
  float tot2 = 0.f;
#pragma unroll
  for (int k = 0; k < 8; ++k) tot2 += part[k];
  float rstd = rsqrtf(tot2 * (1.0f / KD) + LN_EPS);

  out[(size_t)rowId * KD + t] = d * rstd * g[t] + be[t];
}

// ---------------------------------------------------------------------------
extern "C" void kernel_launch(void* const* d_in, const int* in_sizes, int n_in,
                              void* d_out, int out_size, void* d_ws, size_t ws_size,
                              hipStream_t stream) {
  const float* x    = (const float*)d_in[0];
  const int*   mask = (const int*)d_in[1];
  const float* W1   = (const float*)d_in[2];
  const float* b1   = (const float*)d_in[3];
  const float* W2   = (const float*)d_in[4];
  const float* b2   = (const float*)d_in[5];
  const float* g1   = (const float*)d_in[6];
  const float* be1  = (const float*)d_in[7];
  const float* g2   = (const float*)d_in[8];
  const float* be2  = (const float*)d_in[9];
  float* out = (float*)d_out;

  // Workspace layout (floats). Total ~64 MB (fits comfortably in 192 MB L2).
  float* ws  = (float*)d_ws;
  float* m_  = ws;                                 //   L*32        =   65,536
  float* E   = m_ + (size_t)KL * KDK;              //   L*L         = 4,194,304
  float* rs  = E + (size_t)KL * KL;                //   B*L         =     8,192
  float* O   = rs + (size_t)KB * KL;               //   B*L*D       = 2,097,152
  float* h   = O + (size_t)KB * KL * KD;           //   B*L*D       = 2,097,152
  float* hid = h + (size_t)KB * KL * KD;           //   B*L*F       = 8,388,608
  float* f2  = O;    // O is dead after ln1 -> reuse for FFN output
  float* xm  = hid;  // xm is dead before ffn1 writes hid -> share storage

  qat_mean<<<dim3(KL * KDK / 256), dim3(256), 0, stream>>>(x, m_);
  qat_maskx<<<dim3(KB * KL * KD / 256), dim3(256), 0, stream>>>(x, mask, xm);
  qat_scores<<<dim3(KL / 16, KL / 128), dim3(256), 0, stream>>>(m_, E);
  qat_rowsum<<<dim3(KB * KL), dim3(256), 0, stream>>>(E, mask, rs);
  qat_attn_gemm<<<dim3(KL / 64, KD / 128, KB), dim3(256), 0, stream>>>(E, xm, O);
  qat_ln1<<<dim3(KB * KL), dim3(256), 0, stream>>>(x, O, rs, g1, be1, h);
  qat_ffn1<<<dim3(KB * KL / 32, KF / 256), dim3(256), 0, stream>>>(h, W1, b1, hid);
  qat_ffn2<<<dim3(KB * KL / 32, KD / 256), dim3(256), 0, stream>>>(hid, W2, b2, f2);
  qat_ln2<<<dim3(KB * KL), dim3(256), 0, stream>>>(h, f2, g2, be2, out);
}